// EqSoNN_25357486916197
// MI455X (gfx1250) — compile-verified
//
#include <hip/hip_runtime.h>
#include <hip/hip_bf16.h>
#include <math.h>

typedef __attribute__((ext_vector_type(2))) float v2f;
typedef __attribute__((ext_vector_type(4))) float v4f;
typedef __attribute__((ext_vector_type(8))) float v8f;

// Problem constants (match the reference)
constexpr int MM     = 41;   // symbols
constexpr int CENTER = 20;   // (M-1)/2
constexpr int HD     = 10;   // HDIM
constexpr int NCOLS  = 48;   // 3 WMMA chains * 16 columns
// Column layout (n in [0,48)):
//   n < 40  : conv columns, n = o + 10*(c + 2*f)   o=hidden, c=re/im, f=normal/flip
//   40..43  : pbc columns,  n = 40 + 2*p + c       p=pol, c=re/im
//   44..47  : zero padding
constexpr int WAVES_PER_BLOCK = 2;
constexpr int ROWS_PER_WAVE   = 16;

// ---------------------------------------------------------------------------
// Prep kernel: repack complex weights into the WMMA B-operand layout.
// bp[s][h][n] is a float2 holding B[K=4s+2h][n], B[K=4s+2h+1][n]
// (h = lane>>4 selects the K-half, per the 16x16x4 f32 B VGPR layout).
// K = 4s + 2i + cc, with i=pol of feature, cc=0 -> Fr coefficient, cc=1 -> Fi.
// Complex product w*F: re = wr*Fr - wi*Fi ; im = wi*Fr + wr*Fi.
// ---------------------------------------------------------------------------
__global__ void eq_prep_kernel(const float* __restrict__ W1,
                               const float* __restrict__ W2,
                               const float* __restrict__ pbcC,
                               float* __restrict__ bp1,
                               float* __restrict__ bp2,
                               int S) {
  int total1 = S * 2 * NCOLS;
  int total2 = MM * 2 * NCOLS;
  for (int t = blockIdx.x * blockDim.x + threadIdx.x;
       t < total1 + total2;
       t += gridDim.x * blockDim.x) {
    if (t < total1) {
      int n = t % NCOLS;
      int rest = t / NCOLS;
      int h = rest & 1;       // = pol index i of the feature component
      int s = rest >> 1;
      float v0 = 0.f, v1 = 0.f;
      if (n < 40) {
        int f = n / 20, rem = n % 20, c = rem / 10, o = rem % 10;
        int iw = f ? (1 - h) : h;                   // pol-flip applies to weights
        const float* w = W1 + ((size_t)(o * 2 + iw) * S + s) * 2;
        float wr = w[0], wi = w[1];
        v0 = c ? wi : wr;                           // coeff of Fr
        v1 = c ? wr : -wi;                          // coeff of Fi
      } else if (n < 44) {
        int p = (n - 40) >> 1, c = (n - 40) & 1;
        if (p == h) {
          float Cr = pbcC[s * 2], Ci = pbcC[s * 2 + 1];
          v0 = c ? Ci : Cr;
          v1 = c ? Cr : -Ci;
        }
      }
      bp1[(size_t)t * 2]     = v0;
      bp1[(size_t)t * 2 + 1] = v1;
    } else {
      int u = t - total1;
      int n = u % NCOLS;
      int rest = u / NCOLS;
      int h = rest & 1;
      int mm = rest >> 1;
      float v0 = 0.f, v1 = 0.f;
      if (n < 40) {
        int f = n / 20, rem = n % 20, c = rem / 10, o = rem % 10;
        int iw = f ? (1 - h) : h;
        const float* w = W2 + ((size_t)(o * 2 + iw) * MM + mm) * 2;
        float wr = w[0], wi = w[1];
        v0 = c ? wi : wr;
        v1 = c ? wr : -wi;
      }
      bp2[(size_t)u * 2]     = v0;
      bp2[(size_t)u * 2 + 1] = v1;
    }
  }
}

// ---------------------------------------------------------------------------
// Main kernel: one wave handles 16 batch rows.
//   Phase 1: stage E tile into LDS (contiguous 16*41*4 floats).
//   Phase 2: triplet-feature GEMM via v_wmma_f32_16x16x4_f32, 3 chains of 16
//            output columns (40 conv cols + 4 pbc cols + padding).
//   Phase 3: raw-E GEMM (W2) with identical structure over M=41.
//   Phase 4: stage D tiles to LDS, per-(b,p) combine: pbc + cubic term.
// ---------------------------------------------------------------------------
__global__ void __launch_bounds__(WAVES_PER_BLOCK * 32)
eq_main_kernel(const float* __restrict__ x,
               const float* __restrict__ task,
               const float* __restrict__ b1,
               const float* __restrict__ b2,
               const int*   __restrict__ m_idx,
               const int*   __restrict__ n_idx,
               const float* __restrict__ bp1,
               const float* __restrict__ bp2,
               float* __restrict__ out,
               int S, int Btotal) {
  __shared__ __align__(16) float Elds[WAVES_PER_BLOCK][ROWS_PER_WAVE * MM * 4];
  __shared__ __align__(16) float Dlds[WAVES_PER_BLOCK][2][3 * 256];

  const int lane = threadIdx.x & 31;
  const int wave = threadIdx.x >> 5;
  const int tileBase = (blockIdx.x * WAVES_PER_BLOCK + wave) * ROWS_PER_WAVE;

  // ---- Phase 1: stage E (x layout [B][M][pol][re/im] is already the LDS layout)
  float* eW = Elds[wave];
  {
    for (int t = lane; t < ROWS_PER_WAVE * MM; t += 32) {
      int row = t / MM;
      int q   = t - row * MM;
      int gb  = tileBase + row;
      if (gb >= Btotal) gb = Btotal - 1;              // clamp (tail safety)
      const v4f v = *(const v4f*)(x + ((size_t)gb * MM + q) * 4);
      *(v4f*)(eW + t * 4) = v;
    }
  }
  __syncthreads();

  const int b = lane & 15;   // batch row within tile (WMMA M / D column-group row)
  const int h = lane >> 4;   // K-half selector == feature pol index
  const float* erow = eW + b * (MM * 4);

  // ---- Phase 2: feature GEMM
  v8f accA0 = {}, accA1 = {}, accA2 = {};
  for (int s = 0; s < S; ++s) {
    int m = m_idx[s];
    int n = n_idx[s];
    const float* pm  = erow + (CENTER + m) * 4;
    const float* pmn = erow + (CENTER + m + n) * 4;
    const float* pn  = erow + (CENTER + n) * 4 + h * 2;
    float am0r = pm[0],  am0i = pm[1],  am1r = pm[2],  am1i = pm[3];
    float an0r = pmn[0], an0i = pmn[1], an1r = pmn[2], an1i = pmn[3];
    // G = sum_p E[c+m,p] * conj(E[c+m+n,p])
    float Gr = am0r * an0r + am0i * an0i + am1r * an1r + am1i * an1i;
    float Gi = am0i * an0r - am0r * an0i + am1i * an1r - am1r * an1i;
    float enr = pn[0], eni = pn[1];
    v2f a;
    a.x = Gr * enr - Gi * eni;   // Fr  (K = 4s + 2h + 0)
    a.y = Gr * eni + Gi * enr;   // Fi  (K = 4s + 2h + 1)

    const float* bb = bp1 + (((size_t)s * 2 + h) * NCOLS + b) * 2;
    v2f w0 = *(const v2f*)(bb);
    v2f w1 = *(const v2f*)(bb + 32);
    v2f w2 = *(const v2f*)(bb + 64);
    accA0 = __builtin_amdgcn_wmma_f32_16x16x4_f32(false, a, false, w0, (short)0, accA0, false, false);
    accA1 = __builtin_amdgcn_wmma_f32_16x16x4_f32(false, a, false, w1, (short)0, accA1, false, false);
    accA2 = __builtin_amdgcn_wmma_f32_16x16x4_f32(false, a, false, w2, (short)0, accA2, false, false);
  }

  // ---- Phase 3: raw-E GEMM (Bm)
  v8f accE0 = {}, accE1 = {}, accE2 = {};
  for (int mm = 0; mm < MM; ++mm) {
    const float* pe = erow + mm * 4 + h * 2;
    v2f a;
    a.x = pe[0];
    a.y = pe[1];
    const float* bb = bp2 + (((size_t)mm * 2 + h) * NCOLS + b) * 2;
    v2f w0 = *(const v2f*)(bb);
    v2f w1 = *(const v2f*)(bb + 32);
    v2f w2 = *(const v2f*)(bb + 64);
    accE0 = __builtin_amdgcn_wmma_f32_16x16x4_f32(false, a, false, w0, (short)0, accE0, false, false);
    accE1 = __builtin_amdgcn_wmma_f32_16x16x4_f32(false, a, false, w1, (short)0, accE1, false, false);
    accE2 = __builtin_amdgcn_wmma_f32_16x16x4_f32(false, a, false, w2, (short)0, accE2, false, false);
  }

  // ---- Phase 4: stage D tiles (C/D layout: VGPR v, lanes<16 -> M=v, lanes>=16 -> M=v+8)
  float* dA = Dlds[wave][0];
  float* dE = Dlds[wave][1];
#pragma unroll
  for (int v = 0; v < 8; ++v) {
    int row = v + 8 * h;
    dA[0 * 256 + row * 16 + b] = accA0[v];
    dA[1 * 256 + row * 16 + b] = accA1[v];
    dA[2 * 256 + row * 16 + b] = accA2[v];
    dE[0 * 256 + row * 16 + b] = accE0[v];
    dE[1 * 256 + row * 16 + b] = accE1[v];
    dE[2 * 256 + row * 16 + b] = accE2[v];
  }
  __syncthreads();

  // ---- Combine: lane handles (batch row b, pol p = h)
  const int p  = h;
  int gb = tileBase + b;
  int gbc = gb < Btotal ? gb : (Btotal - 1);

  float P = exp10f(task[(size_t)gbc * 4] * 0.1f) * 0.5f;   // 10^(dBm/10)/NMODES

  // center symbol (E still resident in LDS)
  float ecr = erow[CENTER * 4 + p * 2 + 0];
  float eci = erow[CENTER * 4 + p * 2 + 1];

  // pbc dot columns
  auto getA = [&](int n) { return dA[(n >> 4) * 256 + b * 16 + (n & 15)]; };
  auto getE = [&](int n) { return dE[(n >> 4) * 256 + b * 16 + (n & 15)]; };

  float outr = ecr + P * getA(40 + 2 * p);
  float outi = eci + P * getA(41 + 2 * p);

  float tr = 0.f, ti = 0.f;
#pragma unroll
  for (int o = 0; o < HD; ++o) {
    float Ar = getA(o + 10 * (0 + 2 * p)) + b1[o * 2];
    float Ai = getA(o + 10 * (1 + 2 * p)) + b1[o * 2 + 1];
    float Br = getE(o + 10 * (0 + 2 * p)) + b2[o * 2];
    float Bi = getE(o + 10 * (1 + 2 * p)) + b2[o * 2 + 1];
    float mag = Br * Br + Bi * Bi;         // Bm * conj(Bm)
    float B2r = Br * Br - Bi * Bi;         // Bm * Bm
    float B2i = 2.f * Br * Bi;
    tr += Ar * mag + (Ar * B2r + Ai * B2i);   // A*|Bm|^2 + conj(A)*Bm^2 (re)
    ti += Ai * mag + (Ar * B2i - Ai * B2r);   //                          (im)
  }
  float scale = (1e-4f / sqrtf((float)HD)) * P * P;
  outr += scale * tr;
  outi += scale * ti;

  if (gb < Btotal) {
    out[((size_t)gb * 2 + p) * 2 + 0] = outr;
    out[((size_t)gb * 2 + p) * 2 + 1] = outi;
  }
}

// ---------------------------------------------------------------------------
extern "C" void kernel_launch(void* const* d_in, const int* in_sizes, int n_in,
                              void* d_out, int out_size, void* d_ws, size_t ws_size,
                              hipStream_t stream) {
  const float* x    = (const float*)d_in[0];
  const float* task = (const float*)d_in[1];
  const float* pbcC = (const float*)d_in[2];
  const float* W1   = (const float*)d_in[3];
  const float* b1   = (const float*)d_in[4];
  const float* W2   = (const float*)d_in[5];
  const float* b2   = (const float*)d_in[6];
  const int*   mi   = (const int*)d_in[7];
  const int*   ni   = (const int*)d_in[8];

  const int S = in_sizes[7];                 // number of triplets
  const int B = in_sizes[0] / (MM * 4);      // batch size

  float* ws  = (float*)d_ws;
  float* bp1 = ws;                                     // S*2*48 float2
  float* bp2 = ws + (size_t)S * 2 * NCOLS * 2;         // 41*2*48 float2

  int prepTotal = (S + MM) * 2 * NCOLS;
  int prepBlocks = (prepTotal + 255) / 256;
  eq_prep_kernel<<<prepBlocks, 256, 0, stream>>>(W1, W2, pbcC, bp1, bp2, S);

  int rowsPerBlock = WAVES_PER_BLOCK * ROWS_PER_WAVE;
  int mainBlocks = (B + rowsPerBlock - 1) / rowsPerBlock;
  eq_main_kernel<<<mainBlocks, WAVES_PER_BLOCK * 32, 0, stream>>>(
      x, task, b1, b2, mi, ni, bp1, bp2, (float*)d_out, S, B);
}